// IterNorm_Single_16527034155453
// MI455X (gfx1250) — compile-verified
//
#include <hip/hip_runtime.h>

// ---------------------------------------------------------------------------
// IterNorm (training branch) for X[64,512,32,32] fp32 on gfx1250 (MI455X).
// d = 512 channels, m = 64*32*32 = 65536 samples.
// All matrix math runs through V_WMMA_F32_16X16X4_F32 (fp32 matrix pipe).
// ---------------------------------------------------------------------------

typedef float v2f __attribute__((ext_vector_type(2)));
typedef float v8f __attribute__((ext_vector_type(8)));

#define CDIM   512
#define MTOT   65536
#define EPSV   1e-5f
#define LDA_PAD 36    // row stride (floats) for [row][k] LDS panels: 16B aligned rows,
                      // 36*m mod 64 banks distinct for m=0..15 -> conflict-free column reads
#define LDB_PAD 132   // row stride for [k][n] LDS panels: 132*4B = 528B, 16B aligned

// x[c, j] lives at X[(j>>10)<<19 + c*1024 + (j&1023)]  (NCHW, j = n*1024 + h*32 + w)

__device__ __forceinline__ v8f wmma_f32(v2f a, v2f b, v8f c) {
  // D = A(16x4) * B(4x16) + C(16x16), fp32 everywhere
  return __builtin_amdgcn_wmma_f32_16x16x4_f32(
      /*neg_a=*/false, a, /*neg_b=*/false, b,
      /*c_mod=*/(short)0, c, /*reuse_a=*/false, /*reuse_b=*/false);
}

// ---------------------------------------------------------------------------
// 1) per-channel mean (deterministic LDS tree reduction, one block per channel)
// ---------------------------------------------------------------------------
__global__ __launch_bounds__(256) void k_mean(const float* __restrict__ X,
                                              float* __restrict__ mean) {
  const int c = blockIdx.x;
  const int t = threadIdx.x;
  float s = 0.f;
  for (int k = t; k < MTOT; k += 256) {
    const int n = k >> 10;
    s += X[(n << 19) + c * 1024 + (k & 1023)];
  }
  __shared__ float red[256];
  red[t] = s;
  __syncthreads();
  for (int w = 128; w > 0; w >>= 1) {
    if (t < w) red[t] += red[t + w];
    __syncthreads();
  }
  if (t == 0) mean[c] = red[0] * (1.0f / (float)MTOT);
}

// ---------------------------------------------------------------------------
// 2) Gram partials: part[chunk] += x_tile * x_tile^T over a 4096-wide K chunk.
//    Grid = (10 upper-triangular 128x128 tiles) x (16 K chunks).
//    8 waves / block, each wave owns a 64x32 sub-tile (4x2 WMMA tiles).
// ---------------------------------------------------------------------------
__global__ __launch_bounds__(256) void k_gram(const float* __restrict__ X,
                                              float* __restrict__ part) {
  __shared__ float ldsA[128 * LDA_PAD];  // [c-row][k]
  __shared__ float ldsB[128 * LDA_PAD];  // [c-col][k]

  // map 0..9 -> upper-triangular (ti,tj), ti<=tj
  const int tile = blockIdx.x;
  int ti, tj;
  if (tile < 4)      { ti = 0; tj = tile; }
  else if (tile < 7) { ti = 1; tj = tile - 3; }
  else if (tile < 9) { ti = 2; tj = tile - 5; }
  else               { ti = 3; tj = 3; }
  const int c0 = ti * 128;
  const int c1 = tj * 128;

  const int t = threadIdx.x, lane = t & 31, wave = t >> 5;
  const int mbase = (wave >> 2) * 64;
  const int nbase = (wave & 3) * 32;

  v8f acc[8];
  {
    v8f z = {0.f, 0.f, 0.f, 0.f, 0.f, 0.f, 0.f, 0.f};
#pragma unroll
    for (int i = 0; i < 8; ++i) acc[i] = z;
  }

  const int r   = t >> 1;          // 0..127  (panel row)
  const int off = (t & 1) * 16;    // 16-float half-row
  const int lm  = lane & 15;
  const int ko  = (lane >> 4) << 1;

  const int kstart = blockIdx.y * 4096;
  for (int k0 = kstart; k0 < kstart + 4096; k0 += 32) {
    const int n = k0 >> 10;
    const int s = (k0 & 1023) + off;  // 32-chunk never crosses a 1024 row
    const float4* ga = (const float4*)(X + (n << 19) + (c0 + r) * 1024 + s);
    const float4* gb = (const float4*)(X + (n << 19) + (c1 + r) * 1024 + s);
    float4 a0 = ga[0], a1 = ga[1], a2 = ga[2], a3 = ga[3];
    float4 b0 = gb[0], b1 = gb[1], b2 = gb[2], b3 = gb[3];
    __syncthreads();
    float4* da = (float4*)(ldsA + r * LDA_PAD + off);
    float4* db = (float4*)(ldsB + r * LDA_PAD + off);
    da[0] = a0; da[1] = a1; da[2] = a2; da[3] = a3;
    db[0] = b0; db[1] = b1; db[2] = b2; db[3] = b3;
    __syncthreads();

#pragma unroll
    for (int kk = 0; kk < 32; kk += 4) {
      v2f a[4];
#pragma unroll
      for (int mi = 0; mi < 4; ++mi) {
        const float* p = ldsA + (mbase + mi * 16 + lm) * LDA_PAD + kk + ko;
        a[mi].x = p[0]; a[mi].y = p[1];
      }
      v2f b[2];
#pragma unroll
      for (int ni = 0; ni < 2; ++ni) {
        const float* p = ldsB + (nbase + ni * 16 + lm) * LDA_PAD + kk + ko;
        b[ni].x = p[0]; b[ni].y = p[1];
      }
#pragma unroll
      for (int mi = 0; mi < 4; ++mi)
#pragma unroll
        for (int ni = 0; ni < 2; ++ni)
          acc[mi * 2 + ni] = wmma_f32(a[mi], b[ni], acc[mi * 2 + ni]);
    }
  }

  float* dst = part + blockIdx.y * (CDIM * CDIM);
  const int lh = (lane >> 4) << 3;
#pragma unroll
  for (int mi = 0; mi < 4; ++mi) {
#pragma unroll
    for (int ni = 0; ni < 2; ++ni) {
      const int i0 = c0 + mbase + mi * 16 + lh;
      const int j  = c1 + nbase + ni * 16 + lm;
#pragma unroll
      for (int v = 0; v < 8; ++v)
        dst[(i0 + v) * CDIM + j] = acc[mi * 2 + ni][v];
    }
  }
}

// ---------------------------------------------------------------------------
// 3) Sigma = sum_chunk(part)/m - mean*mean^T + eps*I  (mirror the triangle)
// ---------------------------------------------------------------------------
__global__ __launch_bounds__(256) void k_assemble(const float* __restrict__ part,
                                                  const float* __restrict__ mean,
                                                  float* __restrict__ Sigma) {
  const int id = blockIdx.x * 256 + threadIdx.x;  // 0..262143
  const int i = id >> 9, j = id & 511;
  const int src = (i <= j) ? (i * CDIM + j) : (j * CDIM + i);  // Gram is symmetric
  float s = 0.f;
#pragma unroll
  for (int ch = 0; ch < 16; ++ch) s += part[ch * (CDIM * CDIM) + src];
  float v = s * (1.0f / (float)MTOT) - mean[i] * mean[j];
  if (i == j) v += EPSV;
  Sigma[id] = v;
}

// 4) rTr = 1/trace(Sigma); also stash sqrt(rTr)
__global__ __launch_bounds__(512) void k_trace(const float* __restrict__ Sigma,
                                               float* __restrict__ scal) {
  const int t = threadIdx.x;
  __shared__ float red[512];
  red[t] = Sigma[t * CDIM + t];
  __syncthreads();
  for (int w = 256; w > 0; w >>= 1) {
    if (t < w) red[t] += red[t + w];
    __syncthreads();
  }
  if (t == 0) {
    const float rTr = 1.0f / red[0];
    scal[0] = rTr;
    scal[1] = sqrtf(rTr);
  }
}

// 5) SigmaN = Sigma * rTr; P = I
__global__ __launch_bounds__(256) void k_scale_init(const float* __restrict__ Sigma,
                                                    const float* __restrict__ scal,
                                                    float* __restrict__ SigmaN,
                                                    float* __restrict__ P) {
  const int id = blockIdx.x * 256 + threadIdx.x;
  const float rTr = scal[0];
  SigmaN[id] = Sigma[id] * rTr;
  P[id] = ((id >> 9) == (id & 511)) ? 1.0f : 0.0f;
}

// ---------------------------------------------------------------------------
// 6) Generic 512x512x512 WMMA GEMM:  Dm = alpha*A@B (+ beta*C if C != null)
// ---------------------------------------------------------------------------
__global__ __launch_bounds__(256) void k_gemm512(const float* __restrict__ A,
                                                 const float* __restrict__ B,
                                                 const float* __restrict__ C,
                                                 float* __restrict__ Dm,
                                                 float alpha, float beta) {
  __shared__ float ldsA[128 * LDA_PAD];  // [m][k]
  __shared__ float ldsB[32 * LDB_PAD];   // [k][n]
  const int r0 = blockIdx.x * 128, c0 = blockIdx.y * 128;
  const int t = threadIdx.x, lane = t & 31, wave = t >> 5;
  const int mbase = (wave >> 2) * 64, nbase = (wave & 3) * 32;

  v8f acc[8];
  {
    v8f z = {0.f, 0.f, 0.f, 0.f, 0.f, 0.f, 0.f, 0.f};
#pragma unroll
    for (int i = 0; i < 8; ++i) acc[i] = z;
  }

  const int ra = t >> 1, offa = (t & 1) * 16;   // A: 128 rows x 32 k
  const int rb = t >> 3, offb = (t & 7) * 16;   // B: 32 k-rows x 128 n
  const int lm = lane & 15;
  const int ko = (lane >> 4) << 1;

  for (int k0 = 0; k0 < CDIM; k0 += 32) {
    const float4* ga = (const float4*)(A + (r0 + ra) * CDIM + k0 + offa);
    const float4* gb = (const float4*)(B + (k0 + rb) * CDIM + c0 + offb);
    float4 a0 = ga[0], a1 = ga[1], a2 = ga[2], a3 = ga[3];
    float4 b0 = gb[0], b1 = gb[1], b2 = gb[2], b3 = gb[3];
    __syncthreads();
    float4* da = (float4*)(ldsA + ra * LDA_PAD + offa);
    float4* db = (float4*)(ldsB + rb * LDB_PAD + offb);
    da[0] = a0; da[1] = a1; da[2] = a2; da[3] = a3;
    db[0] = b0; db[1] = b1; db[2] = b2; db[3] = b3;
    __syncthreads();

#pragma unroll
    for (int kk = 0; kk < 32; kk += 4) {
      v2f a[4];
#pragma unroll
      for (int mi = 0; mi < 4; ++mi) {
        const float* p = ldsA + (mbase + mi * 16 + lm) * LDA_PAD + kk + ko;
        a[mi].x = p[0]; a[mi].y = p[1];
      }
      v2f b[2];
#pragma unroll
      for (int ni = 0; ni < 2; ++ni) {
        const int col = nbase + ni * 16 + lm;
        b[ni].x = ldsB[(kk + ko) * LDB_PAD + col];
        b[ni].y = ldsB[(kk + ko + 1) * LDB_PAD + col];
      }
#pragma unroll
      for (int mi = 0; mi < 4; ++mi)
#pragma unroll
        for (int ni = 0; ni < 2; ++ni)
          acc[mi * 2 + ni] = wmma_f32(a[mi], b[ni], acc[mi * 2 + ni]);
    }
  }

  const int lh = (lane >> 4) << 3;
#pragma unroll
  for (int mi = 0; mi < 4; ++mi) {
#pragma unroll
    for (int ni = 0; ni < 2; ++ni) {
      const int i0 = r0 + mbase + mi * 16 + lh;
      const int j  = c0 + nbase + ni * 16 + lm;
#pragma unroll
      for (int v = 0; v < 8; ++v) {
        const int idx = (i0 + v) * CDIM + j;
        float val = alpha * acc[mi * 2 + ni][v];
        if (C) val += beta * C[idx];
        Dm[idx] = val;
      }
    }
  }
}

// 7) wb[c] = sqrt(rTr) * dot(P[c,:], mean)
__global__ __launch_bounds__(512) void k_wb(const float* __restrict__ P,
                                            const float* __restrict__ mean,
                                            const float* __restrict__ scal,
                                            float* __restrict__ wb) {
  const int c = threadIdx.x;
  float s = 0.f;
  for (int k = 0; k < CDIM; ++k) s += P[c * CDIM + k] * mean[k];
  wb[c] = scal[1] * s;
}

// ---------------------------------------------------------------------------
// 8) Whitening apply: out[c,j] = sqrt(rTr)*(P @ x)[c,j] - wb[c], NCHW layout.
//    M=512 (4 row tiles) x N=65536 (512 col tiles), K=512.
// ---------------------------------------------------------------------------
__global__ __launch_bounds__(256) void k_whiten(const float* __restrict__ X,
                                                const float* __restrict__ P,
                                                const float* __restrict__ wb,
                                                const float* __restrict__ scal,
                                                float* __restrict__ out) {
  __shared__ float ldsA[128 * LDA_PAD];  // P tile [c][k]
  __shared__ float ldsB[32 * LDB_PAD];   // x tile [k=c'][j]
  const int r0 = blockIdx.x * 128;       // channel tile
  const int j0 = blockIdx.y * 128;       // sample tile (within one 1024-row)
  const int nq = j0 >> 10, sj0 = j0 & 1023;

  const int t = threadIdx.x, lane = t & 31, wave = t >> 5;
  const int mbase = (wave >> 2) * 64, nbase = (wave & 3) * 32;
  const float s1 = scal[1];

  v8f acc[8];
  {
    v8f z = {0.f, 0.f, 0.f, 0.f, 0.f, 0.f, 0.f, 0.f};
#pragma unroll
    for (int i = 0; i < 8; ++i) acc[i] = z;
  }

  const int ra = t >> 1, offa = (t & 1) * 16;
  const int rb = t >> 3, offb = (t & 7) * 16;
  const int lm = lane & 15;
  const int ko = (lane >> 4) << 1;

  for (int k0 = 0; k0 < CDIM; k0 += 32) {
    const float4* ga = (const float4*)(P + (r0 + ra) * CDIM + k0 + offa);
    const float4* gb =
        (const float4*)(X + (nq << 19) + (k0 + rb) * 1024 + sj0 + offb);
    float4 a0 = ga[0], a1 = ga[1], a2 = ga[2], a3 = ga[3];
    float4 b0 = gb[0], b1 = gb[1], b2 = gb[2], b3 = gb[3];
    __syncthreads();
    float4* da = (float4*)(ldsA + ra * LDA_PAD + offa);
    float4* db = (float4*)(ldsB + rb * LDB_PAD + offb);
    da[0] = a0; da[1] = a1; da[2] = a2; da[3] = a3;
    db[0] = b0; db[1] = b1; db[2] = b2; db[3] = b3;
    __syncthreads();

#pragma unroll
    for (int kk = 0; kk < 32; kk += 4) {
      v2f a[4];
#pragma unroll
      for (int mi = 0; mi < 4; ++mi) {
        const float* p = ldsA + (mbase + mi * 16 + lm) * LDA_PAD + kk + ko;
        a[mi].x = p[0]; a[mi].y = p[1];
      }
      v2f b[2];
#pragma unroll
      for (int ni = 0; ni < 2; ++ni) {
        const int col = nbase + ni * 16 + lm;
        b[ni].x = ldsB[(kk + ko) * LDB_PAD + col];
        b[ni].y = ldsB[(kk + ko + 1) * LDB_PAD + col];
      }
#pragma unroll
      for (int mi = 0; mi < 4; ++mi)
#pragma unroll
        for (int ni = 0; ni < 2; ++ni)
          acc[mi * 2 + ni] = wmma_f32(a[mi], b[ni], acc[mi * 2 + ni]);
    }
  }

  // (P@x)[c,j]*s1 - wb[c] -> out at NCHW address (nq<<19) + c*1024 + (j&1023)
  const int lh = (lane >> 4) << 3;
#pragma unroll
  for (int mi = 0; mi < 4; ++mi) {
#pragma unroll
    for (int ni = 0; ni < 2; ++ni) {
      const int c  = r0 + mbase + mi * 16 + lh;
      const int sj = sj0 + nbase + ni * 16 + lm;
#pragma unroll
      for (int v = 0; v < 8; ++v) {
        out[(nq << 19) + (c + v) * 1024 + sj] =
            s1 * acc[mi * 2 + ni][v] - wb[c + v];
      }
    }
  }
}

// ---------------------------------------------------------------------------
extern "C" void kernel_launch(void* const* d_in, const int* in_sizes, int n_in,
                              void* d_out, int out_size, void* d_ws, size_t ws_size,
                              hipStream_t stream) {
  const float* X = (const float*)d_in[0];
  float* out = (float*)d_out;

  // small-matrix scratch in d_ws (~6.3 MB)
  float* ws     = (float*)d_ws;
  float* mean   = ws;                    // 512
  float* wb     = ws + 512;              // 512
  float* scal   = ws + 1024;             // 2 (+pad)
  float* Sigma  = ws + 1040;             // 512*512
  float* SigmaN = Sigma + CDIM * CDIM;
  float* Pa     = SigmaN + CDIM * CDIM;
  float* Pb     = Pa + CDIM * CDIM;
  float* T1     = Pb + CDIM * CDIM;
  float* T2     = T1 + CDIM * CDIM;

  // Gram split-K partials (16 MB) live in d_out: it's dead space until k_whiten
  // overwrites all of it at the end.
  float* part = out;

  k_mean<<<512, 256, 0, stream>>>(X, mean);
  k_gram<<<dim3(10, 16), 256, 0, stream>>>(X, part);
  k_assemble<<<1024, 256, 0, stream>>>(part, mean, Sigma);
  k_trace<<<1, 512, 0, stream>>>(Sigma, scal);
  k_scale_init<<<1024, 256, 0, stream>>>(Sigma, scal, SigmaN, Pa);

  float* P  = Pa;
  float* Pn = Pb;
  for (int it = 0; it < 5; ++it) {
    // T1 = P@P ; T2 = T1@P ; Pn = 1.5*P - 0.5*T2@SigmaN
    k_gemm512<<<dim3(4, 4), 256, 0, stream>>>(P, P, nullptr, T1, 1.0f, 0.0f);
    k_gemm512<<<dim3(4, 4), 256, 0, stream>>>(T1, P, nullptr, T2, 1.0f, 0.0f);
    k_gemm512<<<dim3(4, 4), 256, 0, stream>>>(T2, SigmaN, P, Pn, -0.5f, 1.5f);
    float* tmp = P; P = Pn; Pn = tmp;
  }

  k_wb<<<1, 512, 0, stream>>>(P, mean, scal, wb);
  k_whiten<<<dim3(4, 512), 256, 0, stream>>>(X, P, wb, scal, out);
}